// ChamferDistance_20220706029590
// MI455X (gfx1250) — compile-verified
//
#include <hip/hip_runtime.h>

typedef __attribute__((ext_vector_type(2))) float v2f;
typedef __attribute__((ext_vector_type(8))) float v8f;

#define BATCH 4
#define DFEAT 3
#define TILE  128   // rows (n) and cols (m) covered per workgroup
#define MSUB  (TILE / 16)

__global__ void chamfer_init(unsigned int* __restrict__ mins, int total) {
  int i = blockIdx.x * blockDim.x + threadIdx.x;
  if (i < total) mins[i] = 0x7F800000u;  // +inf bits
}

__global__ __launch_bounds__(256) void chamfer_tile(
    const float* __restrict__ x, const float* __restrict__ y,
    unsigned int* __restrict__ minx, unsigned int* __restrict__ miny,
    int n, int m) {
  __shared__ unsigned int colmin[TILE];

  const int tid  = threadIdx.x;
  const int lane = tid & 31;
  const int wid  = tid >> 5;                       // 8 waves per workgroup
  const int b    = blockIdx.z;
  const int n0   = blockIdx.x * TILE + wid * 16;   // this wave's 16-row strip
  const int m0   = blockIdx.y * TILE;

  if (tid < TILE) colmin[tid] = 0x7F800000u;
  __syncthreads();

  const float* xb = x + (size_t)b * DFEAT * n;
  const float* yb = y + (size_t)b * DFEAT * m;

  const int  half = lane & 15;
  const bool lo   = (lane < 16);
  const int  row  = n0 + half;

  // A-matrix 16x4 f32, row r = [x0, x1, x2, ||x_r||^2]:
  //   lower lanes hold K={0,1}, upper lanes hold K={2,3}
  float x0 = xb[row];
  float x1 = xb[n + row];
  float x2 = xb[2 * n + row];
  float xxl = x0 * x0 + x1 * x1 + x2 * x2;
  v2f a;
  a.x = lo ? x0 : x2;
  a.y = lo ? x1 : xxl;

  float rmin[8];
#pragma unroll
  for (int j = 0; j < 8; ++j) rmin[j] = __builtin_inff();

#pragma unroll
  for (int it = 0; it < MSUB; ++it) {
    const int col = m0 + it * 16 + half;
    float y0 = yb[col];
    float y1 = yb[m + col];
    float y2 = yb[2 * m + col];
    float yy = y0 * y0 + y1 * y1 + y2 * y2;  // per-lane ||y_col||^2

    // B-matrix 4x16, col c = [-2y0, -2y1, -2y2, 1]
    v2f bv;
    bv.x = lo ? (-2.0f * y0) : (-2.0f * y2);
    bv.y = lo ? (-2.0f * y1) : 1.0f;

    v8f c = {};
    // c[j] = ||x_row||^2 - 2 * (x_row . y_col)   (xx folded via K=3 slot)
    c = __builtin_amdgcn_wmma_f32_16x16x4_f32(
        /*neg_a=*/false, a, /*neg_b=*/false, bv,
        /*c_mod=*/(short)0, c, /*reuse_a=*/false, /*reuse_b=*/false);

    float cmin = __builtin_inff();
#pragma unroll
    for (int j = 0; j < 8; ++j) {
      float d2 = fmaxf(c[j] + yy, 0.0f);  // clamp: true distance >= 0
      rmin[j] = fminf(rmin[j], d2);
      cmin    = fminf(cmin, d2);
    }
    // column min over the full 16 rows: combine the two 8-row halves.
    // Both halves then hold the same value for the same (lane&15) column;
    // the duplicated LDS atomic-min is idempotent, so no divergent branch.
    cmin = fminf(cmin, __shfl_xor(cmin, 16, 32));
    atomicMin(&colmin[it * 16 + half], __float_as_uint(cmin));
  }

  // row mins: reduce over the 16 lanes of each half (N dimension).
  // After the xor tree, every lane of the lower half holds row (n0+j)'s min
  // and every lane of the upper half holds row (n0+8+j)'s min.
  const int rofs = (lane & 16) ? 8 : 0;
#pragma unroll
  for (int j = 0; j < 8; ++j) {
    float v = rmin[j];
    v = fminf(v, __shfl_xor(v, 1, 32));
    v = fminf(v, __shfl_xor(v, 2, 32));
    v = fminf(v, __shfl_xor(v, 4, 32));
    v = fminf(v, __shfl_xor(v, 8, 32));
    if (half == 0)
      atomicMin(&minx[(size_t)b * n + n0 + rofs + j], __float_as_uint(v));
  }

  __syncthreads();
  if (tid < TILE) atomicMin(&miny[(size_t)b * m + m0 + tid], colmin[tid]);
}

// Deterministic single-block reduction: fixed per-thread serial order + tree.
__global__ __launch_bounds__(256) void chamfer_reduce(
    const unsigned int* __restrict__ mins, int total, float inv_b,
    float* __restrict__ out) {
  __shared__ float sdata[256];
  float s = 0.0f;
  for (int i = threadIdx.x; i < total; i += 256) s += __uint_as_float(mins[i]);
  sdata[threadIdx.x] = s;
  __syncthreads();
  for (int off = 128; off > 0; off >>= 1) {
    if ((int)threadIdx.x < off) sdata[threadIdx.x] += sdata[threadIdx.x + off];
    __syncthreads();
  }
  if (threadIdx.x == 0) out[0] = sdata[0] * inv_b;
}

extern "C" void kernel_launch(void* const* d_in, const int* in_sizes, int n_in,
                              void* d_out, int out_size, void* d_ws, size_t ws_size,
                              hipStream_t stream) {
  const float* x = (const float*)d_in[0];
  const float* y = (const float*)d_in[1];
  const int n = in_sizes[0] / (BATCH * DFEAT);  // 4096
  const int m = in_sizes[1] / (BATCH * DFEAT);  // 4096

  unsigned int* minx = (unsigned int*)d_ws;
  unsigned int* miny = minx + (size_t)BATCH * n;
  const int total = BATCH * (n + m);

  chamfer_init<<<(total + 255) / 256, 256, 0, stream>>>(minx, total);

  dim3 grid(n / TILE, m / TILE, BATCH);
  chamfer_tile<<<grid, 256, 0, stream>>>(x, y, minx, miny, n, m);

  chamfer_reduce<<<1, 256, 0, stream>>>(minx, total, 1.0f / (float)BATCH,
                                        (float*)d_out);
}